// D2GCN_2448131359460
// MI455X (gfx1250) — compile-verified
//
#include <hip/hip_runtime.h>
#include <hip/hip_bf16.h>

typedef float v2f __attribute__((ext_vector_type(2)));
typedef float v8f __attribute__((ext_vector_type(8)));

#define WMMA4(A, B, C) __builtin_amdgcn_wmma_f32_16x16x4_f32(false, (A), false, (B), (short)0, (C), false, false)

static constexpr float kEps = 1e-5f;

// ---------------- Kernel 1: per-node attention logits + residual init + ws init ----
__global__ void __launch_bounds__(256) k_node_init(
    const float* __restrict__ x, const float* __restrict__ al, const float* __restrict__ ar,
    float* __restrict__ a1, float* __restrict__ a2,
    unsigned* __restrict__ key, float* __restrict__ den,
    float* __restrict__ out, int N)
{
    int gw   = (blockIdx.x * blockDim.x + threadIdx.x) >> 5;   // one wave per node
    int lane = threadIdx.x & 31;
    if (gw >= N) return;
    const float* row  = x   + (long)gw * 256;
    float*       orow = out + (long)gw * 256;
    float s1 = 0.f, s2 = 0.f;
    #pragma unroll
    for (int k = lane; k < 256; k += 32) {
        float v = row[k];
        s1 += v * al[k];
        s2 += v * ar[k];
        orow[k] = v;                       // residual: out starts as n_feature
    }
    for (int off = 16; off; off >>= 1) {
        s1 += __shfl_down(s1, off);
        s2 += __shfl_down(s2, off);
    }
    if (lane == 0) { a1[gw] = s1; a2[gw] = s2; key[gw] = 0u; den[gw] = 0.f; }
}

// ---------------- Kernel 2: edge logits + segment max (ordered-uint atomicMax) -----
__global__ void __launch_bounds__(256) k_edge_logit(
    const float* __restrict__ a1, const float* __restrict__ a2,
    const int* __restrict__ src, const int* __restrict__ dst,
    float* __restrict__ ae, unsigned* __restrict__ key, int E)
{
    int e = blockIdx.x * blockDim.x + threadIdx.x;
    if (e >= E) return;
    float a = a1[src[e]] + a2[dst[e]];
    a = fmaxf(a, 0.f) + 0.2f * fminf(a, 0.f);      // LeakyReLU(0.2)
    ae[e] = a;
    unsigned i = __float_as_uint(a);
    unsigned k = (i & 0x80000000u) ? ~i : (i | 0x80000000u);   // order-preserving encode
    atomicMax(&key[dst[e]], k);
}

// ---------------- Kernel 3: exp(a - amax[dst]) + segment sum ----------------------
__global__ void __launch_bounds__(256) k_edge_exp(
    float* __restrict__ ae, const unsigned* __restrict__ key,
    float* __restrict__ den, const int* __restrict__ dst, int E)
{
    int e = blockIdx.x * blockDim.x + threadIdx.x;
    if (e >= E) return;
    unsigned k = key[dst[e]];
    unsigned i = (k & 0x80000000u) ? (k & 0x7fffffffu) : ~k;   // decode
    float ex = __expf(ae[e] - __uint_as_float(i));
    ae[e] = ex;
    atomicAdd(&den[dst[e]], ex);
}

// ---------------- Kernel 4: fused edge MLP + node MLP + attention + scatter --------
// 16 edges per wave, 2 waves per 64-thread block.
__global__ void __launch_bounds__(64) k_edge_mlp(
    const float* __restrict__ x,
    const int* __restrict__ src, const int* __restrict__ dst,
    const float* feW1, const float* feB1, const float* feG1, const float* feBe1, const float* feM1, const float* feV1,
    const float* feW2, const float* feB2, const float* feG2, const float* feBe2, const float* feM2, const float* feV2,
    const float* fnW1, const float* fnB1, const float* fnG1, const float* fnBe1, const float* fnM1, const float* fnV1,
    const float* fnW2, const float* fnB2, const float* fnG2, const float* fnBe2, const float* fnM2, const float* fnV2,
    const float* __restrict__ ae, const float* __restrict__ den,
    float* __restrict__ out, int E)
{
    __shared__ float lB1[512 * 8];          // cols 0-3: fe_W1[k][n], cols 4-7: fn_W1[k][n]
    __shared__ float lW2e[4 * 256];
    __shared__ float lW2n[4 * 256];
    __shared__ float s2e[256], t2e[256], s2n[256], t2n[256];
    __shared__ float s1f[8], t1f[8];        // folded BN layer1: [0..3]=edge, [4..7]=node
    __shared__ float ef[2][16 * 256];       // per-wave edge_feat staging
    __shared__ float h1b[2][64];            // per-wave 16x4 hidden staging
    __shared__ float la[2][16];             // per-wave attention coeffs
    __shared__ int   ld[2][16];             // per-wave dst indices

    const int tid = threadIdx.x;

    // ---- stage weights + folded BN into LDS ----
    for (int i = tid; i < 2048; i += 64) {
        int k = i >> 2, c = i & 3;
        lB1[k * 8 + c]     = feW1[i];
        lB1[k * 8 + 4 + c] = fnW1[i];
    }
    for (int i = tid; i < 1024; i += 64) { lW2e[i] = feW2[i]; lW2n[i] = fnW2[i]; }
    for (int f = tid; f < 256; f += 64) {
        float s = feG2[f] * rsqrtf(feV2[f] + kEps);
        s2e[f] = s; t2e[f] = (feB2[f] - feM2[f]) * s + feBe2[f];
        s = fnG2[f] * rsqrtf(fnV2[f] + kEps);
        s2n[f] = s; t2n[f] = (fnB2[f] - fnM2[f]) * s + fnBe2[f];
    }
    if (tid < 4) {
        float s = feG1[tid] * rsqrtf(feV1[tid] + kEps);
        s1f[tid] = s; t1f[tid] = (feB1[tid] - feM1[tid]) * s + feBe1[tid];
    } else if (tid < 8) {
        int q = tid - 4;
        float s = fnG1[q] * rsqrtf(fnV1[q] + kEps);
        s1f[tid] = s; t1f[tid] = (fnB1[q] - fnM1[q]) * s + fnBe1[q];
    }
    __syncthreads();

    const int  w    = tid >> 5;
    const int  lane = tid & 31;
    const long tb   = ((long)blockIdx.x * 2 + w) * 16;
    if (tb >= E) return;                               // wave-uniform exit

    const int m  = lane & 15;        // edge row (A matrix M / D row group)
    const int hh = lane >> 4;        // lane half selects K pair
    const int ko = hh * 2;
    const int n  = m;                // B col / D col for this lane
    const int nn = n & 7;

    long e  = tb + m;
    long ec = (e < E) ? e : (long)(E - 1);
    const int si = src[ec];
    const int di = dst[ec];
    const float* rowS = x + (long)si * 256;
    const float* rowD = x + (long)di * 256;

    if (lane < 16) {
        float att = 0.f;
        if (e < E) att = ae[e] / den[di];              // padded edges -> att 0
        la[w][lane] = att;
        ld[w][lane] = di;
    }

    float* EF = ef[w];
    float* H1 = h1b[w];

    v8f acc = {};
    // pass 1: k in [0,256), A = h_src feeds BOTH MLPs (B cols 0-3 fe, 4-7 fn)
    for (int kb = 0; kb < 256; kb += 4) {
        int kr = kb + ko;
        v2f a = *(const v2f*)(rowS + kr);
        v2f b = { lB1[kr * 8 + nn], lB1[(kr + 1) * 8 + nn] };
        acc = WMMA4(a, b, acc);
    }
    // pass 2: k in [256,512), A = h_dst, only fe cols active
    for (int kb = 0; kb < 256; kb += 4) {
        int kr  = kb + ko;
        int krw = 256 + kr;
        v2f a = *(const v2f*)(rowD + kr);
        v2f b;
        b.x = (n < 4) ? lB1[krw * 8 + nn]       : 0.f;
        b.y = (n < 4) ? lB1[(krw + 1) * 8 + nn] : 0.f;
        acc = WMMA4(a, b, acc);
    }
    // edge MLP hidden (cols 0-3): BN + LeakyReLU(0.01) -> LDS in A-frag-friendly layout
    if (n < 4) {
        float s = s1f[n], t = t1f[n];
        #pragma unroll
        for (int j = 0; j < 8; j++) {
            float z = acc[j] * s + t;
            z = fmaxf(z, 0.f) + 0.01f * fminf(z, 0.f);
            H1[(j + hh * 8) * 4 + n] = z;
        }
    }
    // edge MLP layer 2: K=4, 16 N-tiles; BN + ReLU -> edge_feat in LDS
    {
        v2f a = *(const v2f*)&H1[m * 4 + ko];
        #pragma unroll
        for (int nt = 0; nt < 16; nt++) {
            int f = nt * 16 + n;
            v2f b = { lW2e[ko * 256 + f], lW2e[(ko + 1) * 256 + f] };
            v8f z8 = {};
            z8 = WMMA4(a, b, z8);
            float s = s2e[f], t = t2e[f];
            #pragma unroll
            for (int j = 0; j < 8; j++) {
                float z = fmaxf(z8[j] * s + t, 0.f);
                EF[(j + hh * 8) * 256 + f] = z;
            }
        }
    }
    // pass 3: node MLP layer1 second half, A = edge_feat (LDS), only fn cols active
    for (int kb = 0; kb < 256; kb += 4) {
        int kr  = kb + ko;
        int krw = 256 + kr;
        v2f a = *(const v2f*)&EF[m * 256 + kr];
        bool sel = (n >= 4) && (n < 8);
        v2f b;
        b.x = sel ? lB1[krw * 8 + nn]       : 0.f;
        b.y = sel ? lB1[(krw + 1) * 8 + nn] : 0.f;
        acc = WMMA4(a, b, acc);
    }
    // node MLP hidden (cols 4-7): BN + LeakyReLU(0.01)
    if ((n >= 4) && (n < 8)) {
        float s = s1f[n], t = t1f[n];
        #pragma unroll
        for (int j = 0; j < 8; j++) {
            float z = acc[j] * s + t;
            z = fmaxf(z, 0.f) + 0.01f * fminf(z, 0.f);
            H1[(j + hh * 8) * 4 + (n - 4)] = z;
        }
    }
    // node MLP layer 2: BN + ReLU, scale by attention, scatter-add
    {
        v2f a = *(const v2f*)&H1[m * 4 + ko];
        #pragma unroll
        for (int nt = 0; nt < 16; nt++) {
            int f = nt * 16 + n;
            v2f b = { lW2n[ko * 256 + f], lW2n[(ko + 1) * 256 + f] };
            v8f z8 = {};
            z8 = WMMA4(a, b, z8);
            float s = s2n[f], t = t2n[f];
            #pragma unroll
            for (int j = 0; j < 8; j++) {
                int mm = j + hh * 8;
                float z = fmaxf(z8[j] * s + t, 0.f);
                float msg = z * la[w][mm];
                atomicAdd(out + (long)ld[w][mm] * 256 + f, msg);
            }
        }
    }
}

extern "C" void kernel_launch(void* const* d_in, const int* in_sizes, int n_in,
                              void* d_out, int out_size, void* d_ws, size_t ws_size,
                              hipStream_t stream) {
    (void)n_in; (void)out_size; (void)ws_size;
    const float* x    = (const float*)d_in[0];
    const int*   src  = (const int*)d_in[1];
    const int*   dst  = (const int*)d_in[2];
    const float* al   = (const float*)d_in[3];
    const float* ar   = (const float*)d_in[4];
    const float* feW1 = (const float*)d_in[5],  *feB1 = (const float*)d_in[6];
    const float* feG1 = (const float*)d_in[7],  *feBe1= (const float*)d_in[8];
    const float* feM1 = (const float*)d_in[9],  *feV1 = (const float*)d_in[10];
    const float* feW2 = (const float*)d_in[11], *feB2 = (const float*)d_in[12];
    const float* feG2 = (const float*)d_in[13], *feBe2= (const float*)d_in[14];
    const float* feM2 = (const float*)d_in[15], *feV2 = (const float*)d_in[16];
    const float* fnW1 = (const float*)d_in[17], *fnB1 = (const float*)d_in[18];
    const float* fnG1 = (const float*)d_in[19], *fnBe1= (const float*)d_in[20];
    const float* fnM1 = (const float*)d_in[21], *fnV1 = (const float*)d_in[22];
    const float* fnW2 = (const float*)d_in[23], *fnB2 = (const float*)d_in[24];
    const float* fnG2 = (const float*)d_in[25], *fnBe2= (const float*)d_in[26];
    const float* fnM2 = (const float*)d_in[27], *fnV2 = (const float*)d_in[28];

    const int N = in_sizes[0] / 256;
    const int E = in_sizes[1];

    float*    ws  = (float*)d_ws;
    float*    a1  = ws;                    // N
    float*    a2  = ws + N;                // N
    unsigned* key = (unsigned*)(ws + 2*N); // N
    float*    den = ws + 3*N;              // N
    float*    ae  = ws + 4*N;              // E
    float*    out = (float*)d_out;

    // K1: node logits + residual init + key/den init (one wave per node)
    k_node_init<<<(N + 7) / 8, 256, 0, stream>>>(x, al, ar, a1, a2, key, den, out, N);
    // K2: edge logits + segment max
    k_edge_logit<<<(E + 255) / 256, 256, 0, stream>>>(a1, a2, src, dst, ae, key, E);
    // K3: exp + segment sum
    k_edge_exp<<<(E + 255) / 256, 256, 0, stream>>>(ae, key, den, dst, E);
    // K4: fused MLPs (16 edges/wave, 2 waves/block)
    const int tiles  = (E + 15) / 16;
    const int blocks = (tiles + 1) / 2;
    k_edge_mlp<<<blocks, 64, 0, stream>>>(x, src, dst,
        feW1, feB1, feG1, feBe1, feM1, feV1,
        feW2, feB2, feG2, feBe2, feM2, feV2,
        fnW1, fnB1, fnG1, fnBe1, fnM1, fnV1,
        fnW2, fnB2, fnG2, fnBe2, fnM2, fnV2,
        ae, den, out, E);
}